// GraphStructureTransformer_76046690943372
// MI455X (gfx1250) — compile-verified
//
#include <hip/hip_runtime.h>

// ---------------- problem constants ----------------
#define N_NODES 50000
#define N_EDGES 640000
#define DIM     128
#define HEADS   8
#define DHEAD   16
#define LAYERS  4
#define FFDIM   512
#define DQC     32      // centrality sub-embedding

typedef float v2f __attribute__((ext_vector_type(2)));
typedef float v8f __attribute__((ext_vector_type(8)));

// ---------- ordered-uint mapping for float atomic max ----------
__device__ __forceinline__ unsigned f32_to_ord(float f) {
    unsigned u = __float_as_uint(f);
    return (u & 0x80000000u) ? ~u : (u | 0x80000000u);
}
__device__ __forceinline__ float ord_to_f32(unsigned u) {
    unsigned v = (u & 0x80000000u) ? (u & 0x7FFFFFFFu) : ~u;
    return __uint_as_float(v);
}

// LDS offset of a generic shared-memory pointer (LDS aperture keeps the
// wave-relative offset in addr[31:0] per CDNA5 ISA 10.2 aperture rules).
__device__ __forceinline__ unsigned lds_offset(const void* p) {
    return (unsigned)(unsigned long long)p;
}

// Async global -> LDS copy of 16 bytes (CDNA5 GLOBAL_LOAD_ASYNC_TO_LDS_B128,
// GV addressing, tracked with ASYNCcnt).
__device__ __forceinline__ void async_copy_b128(unsigned lds_dst, const void* gsrc) {
    asm volatile("global_load_async_to_lds_b128 %0, %1, off"
                 :: "v"(lds_dst), "v"(gsrc)
                 : "memory");
}
__device__ __forceinline__ void wait_asynccnt0() {
    asm volatile("s_wait_asynccnt 0x0" ::: "memory");
}

// =====================================================================
// Generic fp32 WMMA GEMM: out[R,C] = A[R,K] @ W[K,C] + bias (+extra) (+relu)
// block = 128 threads (4 waves), tile = 64x64, K staged in LDS chunks of 32
// via async-to-LDS b128 copies. Each wave owns 16 rows x 64 cols =
// 4 accumulators of 16x16, fed by v_wmma_f32_16x16x4_f32.
// =====================================================================
template<bool RELU, bool HAS_EXTRA>
__global__ __launch_bounds__(128)
void gemm_wmma_kernel(const float* __restrict__ A, const float* __restrict__ W,
                      const float* __restrict__ bias, const float* __restrict__ extra,
                      float* __restrict__ out, int R, int K, int C)
{
    // stride 36 floats: 16-byte aligned rows AND conflict-free 16-row column
    // reads (36*r mod 64 distinct for r = 0..15)
    __shared__ float As[64][36];
    __shared__ float Ws[32][64];

    const int tid  = threadIdx.x;
    const int wid  = tid >> 5;
    const int lane = tid & 31;
    const int m    = lane & 15;        // row/col within 16
    const int kg   = lane >> 4;        // 0/1 : which K-pair group
    const int rBase = blockIdx.x * 64;
    const int cBase = blockIdx.y * 64;

    v8f acc[4];
    #pragma unroll
    for (int t = 0; t < 4; ++t)
        #pragma unroll
        for (int i = 0; i < 8; ++i) acc[t][i] = 0.0f;

    for (int k0 = 0; k0 < K; k0 += 32) {
        // ---- stage A tile: 64 rows x 32 k, async global->LDS ----
        #pragma unroll
        for (int i = 0; i < 4; ++i) {
            int idx = tid + i * 128;            // 0..511
            int r   = idx >> 3;                 // 0..63
            int c4  = (idx & 7) * 4;            // 0..28
            int gr  = rBase + r;
            if (gr < R) {
                async_copy_b128(lds_offset(&As[r][c4]),
                                A + (size_t)gr * K + k0 + c4);
            } else {
                *(float4*)&As[r][c4] = make_float4(0.f, 0.f, 0.f, 0.f);
            }
        }
        // ---- stage W tile: 32 k x 64 cols, async global->LDS ----
        #pragma unroll
        for (int i = 0; i < 4; ++i) {
            int idx = tid + i * 128;
            int r   = idx >> 4;                 // 0..31
            int c4  = (idx & 15) * 4;           // 0..60
            async_copy_b128(lds_offset(&Ws[r][c4]),
                            W + (size_t)(k0 + r) * C + cBase + c4);
        }
        wait_asynccnt0();
        __syncthreads();

        #pragma unroll
        for (int kk = 0; kk < 8; ++kk) {
            v2f a;
            a.x = As[wid * 16 + m][kk * 4 + kg * 2 + 0];
            a.y = As[wid * 16 + m][kk * 4 + kg * 2 + 1];
            #pragma unroll
            for (int t = 0; t < 4; ++t) {
                v2f b;
                b.x = Ws[kk * 4 + kg * 2 + 0][t * 16 + m];
                b.y = Ws[kk * 4 + kg * 2 + 1][t * 16 + m];
                acc[t] = __builtin_amdgcn_wmma_f32_16x16x4_f32(
                    false, a, false, b, (short)0, acc[t], false, false);
            }
        }
        __syncthreads();
    }

    // ---- epilogue: C/D layout -> vgpr i holds M = i + 8*kg, N = m ----
    #pragma unroll
    for (int t = 0; t < 4; ++t) {
        int col = cBase + t * 16 + m;
        #pragma unroll
        for (int i = 0; i < 8; ++i) {
            int r = rBase + wid * 16 + kg * 8 + i;
            if (r < R) {
                float v = acc[t][i] + bias[col];
                if (HAS_EXTRA) v += extra[(size_t)r * C + col];
                if (RELU) v = fmaxf(v, 0.0f);
                out[(size_t)r * C + col] = v;
            }
        }
    }
}

// =====================================================================
// Centrality positional encoding: one wave per node, in-wave LayerNorm.
// lane L owns features [4L, 4L+4); lanes 0-7 deg, 8-15 pr, 16-23 cl, 24-31 bt
// =====================================================================
__global__ __launch_bounds__(256)
void centrality_kernel(const int* __restrict__ degree, const float* __restrict__ pagerank,
                       const float* __restrict__ clustering, const float* __restrict__ betweenness,
                       const float* __restrict__ deg_emb,
                       const float* __restrict__ pr_w, const float* __restrict__ pr_b,
                       const float* __restrict__ cl_w, const float* __restrict__ cl_b,
                       const float* __restrict__ bt_w, const float* __restrict__ bt_b,
                       const float* __restrict__ g, const float* __restrict__ b,
                       float* __restrict__ cent, int n)
{
    int node = blockIdx.x * (blockDim.x >> 5) + (threadIdx.x >> 5);
    if (node >= n) return;
    int lane = threadIdx.x & 31;
    int f0   = lane * 4;
    int quad = f0 >> 5;
    int w0   = f0 & 31;

    float vals[4];
    if (quad == 0) {
        int d = degree[node]; d = min(max(d, 0), 999);
        #pragma unroll
        for (int j = 0; j < 4; ++j) vals[j] = deg_emb[d * DQC + w0 + j];
    } else {
        float s; const float* wv; const float* bv;
        if (quad == 1)      { s = pagerank[node];    wv = pr_w; bv = pr_b; }
        else if (quad == 2) { s = clustering[node];  wv = cl_w; bv = cl_b; }
        else                { s = betweenness[node]; wv = bt_w; bv = bt_b; }
        #pragma unroll
        for (int j = 0; j < 4; ++j) vals[j] = s * wv[w0 + j] + bv[w0 + j];
    }

    float sum = vals[0] + vals[1] + vals[2] + vals[3];
    #pragma unroll
    for (int off = 16; off > 0; off >>= 1) sum += __shfl_xor(sum, off, 32);
    float mean = sum * (1.0f / DIM);
    float vsum = 0.f;
    #pragma unroll
    for (int j = 0; j < 4; ++j) { float d = vals[j] - mean; vsum += d * d; }
    #pragma unroll
    for (int off = 16; off > 0; off >>= 1) vsum += __shfl_xor(vsum, off, 32);
    float rstd = rsqrtf(vsum * (1.0f / DIM) + 1e-5f);
    #pragma unroll
    for (int j = 0; j < 4; ++j)
        cent[(size_t)node * DIM + f0 + j] =
            (vals[j] - mean) * rstd * g[f0 + j] + b[f0 + j];
}

// =====================================================================
// Fused (residual-add +) LayerNorm, one wave per row (D = 128)
// =====================================================================
__global__ __launch_bounds__(256)
void layernorm_kernel(const float* __restrict__ base, const float* __restrict__ delta,
                      const float* __restrict__ g, const float* __restrict__ b,
                      float* __restrict__ out, int n)
{
    int node = blockIdx.x * (blockDim.x >> 5) + (threadIdx.x >> 5);
    if (node >= n) return;
    int lane = threadIdx.x & 31;
    int f0   = lane * 4;

    float4 v = *(const float4*)(base + (size_t)node * DIM + f0);
    if (delta) {
        float4 dl = *(const float4*)(delta + (size_t)node * DIM + f0);
        v.x += dl.x; v.y += dl.y; v.z += dl.z; v.w += dl.w;
    }
    float sum = v.x + v.y + v.z + v.w;
    #pragma unroll
    for (int off = 16; off > 0; off >>= 1) sum += __shfl_xor(sum, off, 32);
    float mean = sum * (1.0f / DIM);
    float dx = v.x - mean, dy = v.y - mean, dz = v.z - mean, dw = v.w - mean;
    float vsum = dx * dx + dy * dy + dz * dz + dw * dw;
    #pragma unroll
    for (int off = 16; off > 0; off >>= 1) vsum += __shfl_xor(vsum, off, 32);
    float rstd = rsqrtf(vsum * (1.0f / DIM) + 1e-5f);

    float4 o;
    o.x = dx * rstd * g[f0 + 0] + b[f0 + 0];
    o.y = dy * rstd * g[f0 + 1] + b[f0 + 1];
    o.z = dz * rstd * g[f0 + 2] + b[f0 + 2];
    o.w = dw * rstd * g[f0 + 3] + b[f0 + 3];
    *(float4*)(out + (size_t)node * DIM + f0) = o;
}

// =====================================================================
// Attention edge passes (thread per (edge, head))
// =====================================================================
__global__ void init_attn_kernel(unsigned* __restrict__ smax, float* __restrict__ denom,
                                 float* __restrict__ agg, int nNH, int nND)
{
    int i = blockIdx.x * blockDim.x + threadIdx.x;
    if (i < nNH) { smax[i] = f32_to_ord(-__builtin_inff()); denom[i] = 0.0f; }
    if (i < nND) agg[i] = 0.0f;
}

__global__ __launch_bounds__(256)
void edge_score_kernel(const int* __restrict__ src, const int* __restrict__ dst,
                       const float* __restrict__ q, const float* __restrict__ k,
                       const float* __restrict__ eproj,
                       float* __restrict__ score, unsigned* __restrict__ smax, int nE)
{
    int idx = blockIdx.x * blockDim.x + threadIdx.x;
    if (idx >= nE * HEADS) return;
    int e = idx >> 3, h = idx & 7;
    int s = src[e], d = dst[e];
    const float4* qp = (const float4*)(q     + (size_t)d * DIM + h * DHEAD);
    const float4* kp = (const float4*)(k     + (size_t)s * DIM + h * DHEAD);
    const float4* ep = (const float4*)(eproj + (size_t)e * DIM + h * DHEAD);
    float acc = 0.f;
    #pragma unroll
    for (int i = 0; i < 4; ++i) {
        float4 qv = qp[i], kv = kp[i], ev = ep[i];
        acc += qv.x * (kv.x + ev.x) + qv.y * (kv.y + ev.y)
             + qv.z * (kv.z + ev.z) + qv.w * (kv.w + ev.w);
    }
    acc *= 0.25f;                     // 1/sqrt(16)
    score[idx] = acc;
    atomicMax(&smax[d * HEADS + h], f32_to_ord(acc));
}

__global__ __launch_bounds__(256)
void edge_exp_kernel(const int* __restrict__ dst,
                     const float* __restrict__ score, const unsigned* __restrict__ smax,
                     float* __restrict__ ex, float* __restrict__ denom, int nE)
{
    int idx = blockIdx.x * blockDim.x + threadIdx.x;
    if (idx >= nE * HEADS) return;
    int e = idx >> 3, h = idx & 7;
    int d = dst[e];
    float m = ord_to_f32(smax[d * HEADS + h]);
    float v = expf(score[idx] - m);
    ex[idx] = v;
    atomicAdd(&denom[d * HEADS + h], v);
}

__global__ __launch_bounds__(256)
void edge_agg_kernel(const int* __restrict__ src, const int* __restrict__ dst,
                     const float* __restrict__ ex, const float* __restrict__ denom,
                     const float* __restrict__ v, const float* __restrict__ eproj,
                     float* __restrict__ agg, int nE)
{
    int idx = blockIdx.x * blockDim.x + threadIdx.x;
    if (idx >= nE * HEADS) return;
    int e = idx >> 3, h = idx & 7;
    int s = src[e], d = dst[e];
    float alpha = ex[idx] / (denom[d * HEADS + h] + 1e-9f);
    const float4* vp = (const float4*)(v     + (size_t)s * DIM + h * DHEAD);
    const float4* ep = (const float4*)(eproj + (size_t)e * DIM + h * DHEAD);
    float* ap = agg + (size_t)d * DIM + h * DHEAD;
    #pragma unroll
    for (int i = 0; i < 4; ++i) {
        float4 vv = vp[i], ev = ep[i];
        atomicAdd(ap + i * 4 + 0, alpha * (vv.x + ev.x));
        atomicAdd(ap + i * 4 + 1, alpha * (vv.y + ev.y));
        atomicAdd(ap + i * 4 + 2, alpha * (vv.z + ev.z));
        atomicAdd(ap + i * 4 + 3, alpha * (vv.w + ev.w));
    }
}

// =====================================================================
// host-side orchestration
// =====================================================================
extern "C" void kernel_launch(void* const* d_in, const int* in_sizes, int n_in,
                              void* d_out, int out_size, void* d_ws, size_t ws_size,
                              hipStream_t stream)
{
    const float* node_features = (const float*)d_in[0];
    const int*   edge_index    = (const int*)  d_in[1];
    const float* edge_features = (const float*)d_in[2];
    const int*   degree        = (const int*)  d_in[3];
    const float* pagerank      = (const float*)d_in[4];
    const float* clustering    = (const float*)d_in[5];
    const float* betweenness   = (const float*)d_in[6];
    const float* node_proj_w   = (const float*)d_in[7];
    const float* node_proj_b   = (const float*)d_in[8];
    const float* deg_emb       = (const float*)d_in[9];
    const float* pr_w = (const float*)d_in[10]; const float* pr_b = (const float*)d_in[11];
    const float* cl_w = (const float*)d_in[12]; const float* cl_b = (const float*)d_in[13];
    const float* bt_w = (const float*)d_in[14]; const float* bt_b = (const float*)d_in[15];
    const float* cent_ln_g = (const float*)d_in[16];
    const float* cent_ln_b = (const float*)d_in[17];
    const float* Wq = (const float*)d_in[18]; const float* bq = (const float*)d_in[19];
    const float* Wk = (const float*)d_in[20]; const float* bk = (const float*)d_in[21];
    const float* Wv = (const float*)d_in[22]; const float* bv = (const float*)d_in[23];
    const float* We = (const float*)d_in[24]; const float* be = (const float*)d_in[25];
    const float* Wo = (const float*)d_in[26]; const float* bo = (const float*)d_in[27];
    const float* norm1_g = (const float*)d_in[28]; const float* norm1_b = (const float*)d_in[29];
    const float* norm2_g = (const float*)d_in[30]; const float* norm2_b = (const float*)d_in[31];
    const float* ffn_w1 = (const float*)d_in[32]; const float* ffn_b1 = (const float*)d_in[33];
    const float* ffn_w2 = (const float*)d_in[34]; const float* ffn_b2 = (const float*)d_in[35];
    const float* out_w = (const float*)d_in[36]; const float* out_b = (const float*)d_in[37];
    const float* final_ln_g = (const float*)d_in[38];
    const float* final_ln_b = (const float*)d_in[39];

    const int* src = edge_index;                 // [E]
    const int* dst = edge_index + N_EDGES;       // [E]

    // ---- workspace layout (floats) ----
    float* ws = (float*)d_ws;
    size_t off = 0;
    float* cent  = ws + off; off += (size_t)N_NODES * DIM;
    float* x     = ws + off; off += (size_t)N_NODES * DIM;
    float* qb    = ws + off; off += (size_t)N_NODES * DIM;   // also attended / out_lin
    float* kb    = ws + off; off += (size_t)N_NODES * DIM;   // also h2
    float* vb    = ws + off; off += (size_t)N_NODES * DIM;
    float* agg   = ws + off; off += (size_t)N_NODES * DIM;
    float* h1    = ws + off; off += (size_t)N_NODES * FFDIM;
    float* eproj = ws + off; off += (size_t)N_EDGES * DIM;
    float* score = ws + off; off += (size_t)N_EDGES * HEADS;
    float* exbuf = ws + off; off += (size_t)N_EDGES * HEADS;
    unsigned* smax = (unsigned*)(ws + off); off += (size_t)N_NODES * HEADS;
    float* denom = ws + off; off += (size_t)N_NODES * HEADS;

    const int rowBlkN = (N_NODES + 63) / 64;     // 782
    const int rowBlkE = (N_EDGES + 63) / 64;     // 10000
    const dim3 gN128(rowBlkN, DIM / 64);         // [N,128] output
    const dim3 gN512(rowBlkN, FFDIM / 64);       // [N,512] output
    const dim3 gE128(rowBlkE, DIM / 64);         // [E,128] output
    const int lnBlocks   = (N_NODES + 7) / 8;    // 8 waves per block
    const int edgeBlocks = (N_EDGES * HEADS + 255) / 256;
    const int initBlocks = (N_NODES * DIM + 255) / 256;

    // ---- centrality encoding ----
    hipLaunchKernelGGL(centrality_kernel, dim3(lnBlocks), dim3(256), 0, stream,
        degree, pagerank, clustering, betweenness, deg_emb,
        pr_w, pr_b, cl_w, cl_b, bt_w, bt_b, cent_ln_g, cent_ln_b, cent, N_NODES);

    // ---- x = node_features @ W + b + cent ----
    hipLaunchKernelGGL((gemm_wmma_kernel<false, true>), gN128, dim3(128), 0, stream,
        node_features, node_proj_w, node_proj_b, cent, x, N_NODES, DIM, DIM);

    for (int l = 0; l < LAYERS; ++l) {
        const float* Wq_l = Wq + (size_t)l * DIM * DIM; const float* bq_l = bq + l * DIM;
        const float* Wk_l = Wk + (size_t)l * DIM * DIM; const float* bk_l = bk + l * DIM;
        const float* Wv_l = Wv + (size_t)l * DIM * DIM; const float* bv_l = bv + l * DIM;
        const float* We_l = We + (size_t)l * DIM * DIM; const float* be_l = be + l * DIM;
        const float* Wo_l = Wo + (size_t)l * DIM * DIM; const float* bo_l = bo + l * DIM;
        const float* w1_l = ffn_w1 + (size_t)l * DIM * FFDIM; const float* b1_l = ffn_b1 + l * FFDIM;
        const float* w2_l = ffn_w2 + (size_t)l * FFDIM * DIM; const float* b2_l = ffn_b2 + l * DIM;

        hipLaunchKernelGGL((gemm_wmma_kernel<false, false>), gN128, dim3(128), 0, stream,
            x, Wq_l, bq_l, nullptr, qb, N_NODES, DIM, DIM);
        hipLaunchKernelGGL((gemm_wmma_kernel<false, false>), gN128, dim3(128), 0, stream,
            x, Wk_l, bk_l, nullptr, kb, N_NODES, DIM, DIM);
        hipLaunchKernelGGL((gemm_wmma_kernel<false, false>), gN128, dim3(128), 0, stream,
            x, Wv_l, bv_l, nullptr, vb, N_NODES, DIM, DIM);
        hipLaunchKernelGGL((gemm_wmma_kernel<false, false>), gE128, dim3(128), 0, stream,
            edge_features, We_l, be_l, nullptr, eproj, N_EDGES, DIM, DIM);

        hipLaunchKernelGGL(init_attn_kernel, dim3(initBlocks), dim3(256), 0, stream,
            smax, denom, agg, N_NODES * HEADS, N_NODES * DIM);
        hipLaunchKernelGGL(edge_score_kernel, dim3(edgeBlocks), dim3(256), 0, stream,
            src, dst, qb, kb, eproj, score, smax, N_EDGES);
        hipLaunchKernelGGL(edge_exp_kernel, dim3(edgeBlocks), dim3(256), 0, stream,
            dst, score, smax, exbuf, denom, N_EDGES);
        hipLaunchKernelGGL(edge_agg_kernel, dim3(edgeBlocks), dim3(256), 0, stream,
            src, dst, exbuf, denom, vb, eproj, agg, N_EDGES);

        // attended = agg @ Wo + bo   (reuse qb)
        hipLaunchKernelGGL((gemm_wmma_kernel<false, false>), gN128, dim3(128), 0, stream,
            agg, Wo_l, bo_l, nullptr, qb, N_NODES, DIM, DIM);
        // x = LN(x + attended)
        hipLaunchKernelGGL(layernorm_kernel, dim3(lnBlocks), dim3(256), 0, stream,
            x, qb, norm1_g + l * DIM, norm1_b + l * DIM, x, N_NODES);
        // h1 = relu(x @ w1 + b1)
        hipLaunchKernelGGL((gemm_wmma_kernel<true, false>), gN512, dim3(128), 0, stream,
            x, w1_l, b1_l, nullptr, h1, N_NODES, DIM, FFDIM);
        // h2 = h1 @ w2 + b2   (reuse kb)
        hipLaunchKernelGGL((gemm_wmma_kernel<false, false>), gN128, dim3(128), 0, stream,
            h1, w2_l, b2_l, nullptr, kb, N_NODES, FFDIM, DIM);
        // x = LN(x + h2)
        hipLaunchKernelGGL(layernorm_kernel, dim3(lnBlocks), dim3(256), 0, stream,
            x, kb, norm2_g + l * DIM, norm2_b + l * DIM, x, N_NODES);
    }

    // ---- output head: LN(x @ out_w + out_b) ----
    hipLaunchKernelGGL((gemm_wmma_kernel<false, false>), gN128, dim3(128), 0, stream,
        x, out_w, out_b, nullptr, qb, N_NODES, DIM, DIM);
    hipLaunchKernelGGL(layernorm_kernel, dim3(lnBlocks), dim3(256), 0, stream,
        qb, nullptr, final_ln_g, final_ln_b, (float*)d_out, N_NODES);
}